// ChainedGNN_19679540150373
// MI455X (gfx1250) — compile-verified
//
#include <hip/hip_runtime.h>
#include <hip/hip_bf16.h>

#define NN   768
#define EE   65536
#define BB   64
#define HST  396   // LDS row stride for big gather buffer (bank-conflict free)
#define H2ST 68    // LDS row stride for 64-wide buffers
#define JKST 196   // LDS row stride for 192-wide buffers

typedef __attribute__((ext_vector_type(2))) float v2f;
typedef __attribute__((ext_vector_type(8))) float v8f;

// ---------------------------------------------------------------------------
// fp32 WMMA 16x16x4 wrapper (D = A(16x4) * B(4x16) + C)
// A layout: lane L (m=L&15): vgpr0=A[m][k0+2*(L>>4)], vgpr1=A[m][k0+2*(L>>4)+1]
// B layout: lane L (n=L&15): vgpr0=B[k0+2*(L>>4)][n], vgpr1=B[k0+2*(L>>4)+1][n]
// C layout: vgpr r, lane L -> (m = r + 8*(L>>4), n = L&15)
// ---------------------------------------------------------------------------
__device__ __forceinline__ v8f wmma4(v2f a, v2f b, v8f c) {
  return __builtin_amdgcn_wmma_f32_16x16x4_f32(false, a, false, b, (short)0, c,
                                               false, false);
}

// Per-wave GEMM of two 16-row tiles (rows ln and ln+16 of a 32-row LDS block)
// against a [K,64] row-major weight matrix (K = 4*ksteps, zero-padded if needed).
__device__ __forceinline__ void gemm_pair(const float* A, int ast,
                                          const float* __restrict__ Wg,
                                          int ksteps, int col, int half, int ln,
                                          v8f& c0, v8f& c1) {
  const float* a0p = A + ln * ast;
  const float* a1p = A + (ln + 16) * ast;
  for (int ks = 0; ks < ksteps; ++ks) {
    const int k0 = ks * 4 + 2 * half;
    v2f b;  b.x  = Wg[k0 * 64 + col];  b.y  = Wg[k0 * 64 + 64 + col];
    v2f a;  a.x  = a0p[k0];            a.y  = a0p[k0 + 1];
    c0 = wmma4(a, b, c0);
    v2f a1; a1.x = a1p[k0];            a1.y = a1p[k0 + 1];
    c1 = wmma4(a1, b, c1);
  }
}

// In-LDS LayerNorm over 32 rows of width K (gamma/beta applied in place).
__device__ __forceinline__ void ln_rows_lds(float* buf, int st, int K,
                                            const float* __restrict__ g,
                                            const float* __restrict__ b,
                                            int tid) {
  __syncthreads();
  if (tid < 32) {
    float* row = buf + tid * st;
    float m = 0.f;
    for (int c = 0; c < K; ++c) m += row[c];
    m /= (float)K;
    float v = 0.f;
    for (int c = 0; c < K; ++c) { float d = row[c] - m; v += d * d; }
    v /= (float)K;
    const float rs = rsqrtf(v + 1e-5f);
    for (int c = 0; c < K; ++c) row[c] = (row[c] - m) * rs * g[c] + b[c];
  }
  __syncthreads();
}

// ---------------------------------------------------------------------------
// Node-feature construction from P0
// ---------------------------------------------------------------------------
__global__ __launch_bounds__(256) void k_rowsum(const float* __restrict__ P0,
                                                float* __restrict__ x) {
  __shared__ float red[256];
  const int b = blockIdx.x, n = blockIdx.y, t = threadIdx.x;
  const float* r = P0 + ((size_t)b * NN + n) * NN;
  float s = r[t] + r[t + 256] + r[t + 512];
  red[t] = s;
  __syncthreads();
  for (int o = 128; o; o >>= 1) { if (t < o) red[t] += red[t + o]; __syncthreads(); }
  if (t == 0) x[n * 132 + b] = red[0];
}

__global__ __launch_bounds__(256) void k_colsum(const float* __restrict__ P0,
                                                float* __restrict__ x) {
  const int b = blockIdx.x, n = blockIdx.y * 256 + threadIdx.x;
  const float* base = P0 + (size_t)b * NN * NN + n;
  float s = 0.f;
  for (int k = 0; k < NN; ++k) s += base[(size_t)k * NN];
  x[n * 132 + 64 + b] = s;
}

__global__ void k_role(const int* __restrict__ tx, const int* __restrict__ rx,
                       float* __restrict__ x) {
  const int n = blockIdx.x * blockDim.x + threadIdx.x;
  if (n >= NN) return;
  const float r0 = (n == tx[0]) ? 1.f : 0.f;
  const float r1 = (n == rx[0]) ? 1.f : 0.f;
  const float r2 = (r0 == 0.f && r1 == 0.f) ? 1.f : 0.f;
  float* p = x + n * 132;
  p[128] = r0; p[129] = r1; p[130] = r2; p[131] = 0.f;
}

// Zero-padded weight copy: [Kt,64] -> [Kp,64]
__global__ void k_padw(const float* __restrict__ Wsrc, int Kt, int Kp,
                       float* __restrict__ Wdst) {
  const int i = blockIdx.x * blockDim.x + threadIdx.x;
  if (i >= Kp * 64) return;
  const int k = i >> 6;
  Wdst[i] = (k < Kt) ? Wsrc[i] : 0.f;
}

__global__ void k_winner(const int* __restrict__ src, const int* __restrict__ dst,
                         int* __restrict__ winner) {
  const int e = blockIdx.x * blockDim.x + threadIdx.x;
  if (e >= EE) return;
  atomicMax(&winner[src[e] * NN + dst[e]], e);
}

// ---------------------------------------------------------------------------
// LayerNorms of inputs
// ---------------------------------------------------------------------------
__global__ void k_ln_x(const float* __restrict__ x, const float* __restrict__ g,
                       const float* __restrict__ b, int Dn,
                       float* __restrict__ xo) {
  const int n = blockIdx.x * blockDim.x + threadIdx.x;
  if (n >= NN) return;
  const float* r = x + n * 132;
  float m = 0.f;
  for (int c = 0; c < Dn; ++c) m += r[c];
  m /= (float)Dn;
  float v = 0.f;
  for (int c = 0; c < Dn; ++c) { float d = r[c] - m; v += d * d; }
  v /= (float)Dn;
  const float rs = rsqrtf(v + 1e-5f);
  float* o = xo + n * 132;
  for (int c = 0; c < Dn; ++c) o[c] = (r[c] - m) * rs * g[c] + b[c];
  for (int c = Dn; c < 132; ++c) o[c] = 0.f;
}

// one wave per edge row; optional relu on input (layers > 0)
__global__ __launch_bounds__(128) void k_ln_e(const float* __restrict__ e_in,
                                              int De, int do_relu,
                                              const float* __restrict__ g,
                                              const float* __restrict__ b,
                                              float* __restrict__ eo) {
  const int wid  = (blockIdx.x * blockDim.x + threadIdx.x) >> 5;
  const int lane = threadIdx.x & 31;
  const float* r = e_in + (size_t)wid * De;
  float vals[4] = {0.f, 0.f, 0.f, 0.f};
  float s = 0.f;
#pragma unroll
  for (int i = 0; i < 4; ++i) {
    const int c = lane + 32 * i;
    if (c < De) {
      float v = r[c];
      if (do_relu) v = fmaxf(v, 0.f);
      vals[i] = v;
      s += v;
    }
  }
  for (int off = 16; off; off >>= 1) s += __shfl_xor(s, off, 32);
  const float m = s / (float)De;
  float vs = 0.f;
#pragma unroll
  for (int i = 0; i < 4; ++i) {
    const int c = lane + 32 * i;
    if (c < De) { float d = vals[i] - m; vs += d * d; }
  }
  for (int off = 16; off; off >>= 1) vs += __shfl_xor(vs, off, 32);
  const float rs = rsqrtf(vs / (float)De + 1e-5f);
  float* o = eo + (size_t)wid * De;
#pragma unroll
  for (int i = 0; i < 4; ++i) {
    const int c = lane + 32 * i;
    if (c < De) o[c] = (vals[i] - m) * rs * g[c] + b[c];
  }
}

// ---------------------------------------------------------------------------
// Fused edge kernel: er1 -> leaky -> er2 -> gate -> (+res) -> epost LN ->
// store es -> FiLM GEMMs + W_msg GEMM -> atomic scatter into x_aggr
// 32 edges per 128-thread block (4 waves, each owns a 16-col tile).
// ---------------------------------------------------------------------------
__global__ __launch_bounds__(128) void k_edge(
    const float* __restrict__ e_ln, int De,
    const float* __restrict__ x_ln, int Dn,
    const int* __restrict__ src, const int* __restrict__ dst,
    const float* __restrict__ W1, int K1steps, const float* __restrict__ b1,
    const float* __restrict__ W2, const float* __restrict__ b2,
    const float* __restrict__ es_prev,
    const float* __restrict__ epost_g, const float* __restrict__ epost_b,
    const float* __restrict__ Ws, const float* __restrict__ bs,
    const float* __restrict__ Wh, const float* __restrict__ bh,
    const float* __restrict__ Wm, int KMsteps,
    float* __restrict__ es_out, float* __restrict__ x_aggr) {
  __shared__ float h[32 * HST];
  __shared__ float h2[32 * H2ST];
  __shared__ int s_src[32], s_dst[32];
  const int tid = threadIdx.x;
  const int e0  = blockIdx.x * 32;
  if (tid < 32) { s_src[tid] = src[e0 + tid]; s_dst[tid] = dst[e0 + tid]; }
  __syncthreads();
  // gather h = [e_ln | x_j | x_i]
  {
    const int r = tid >> 2, q = tid & 3;
    const int e = e0 + r;
    float* hr = h + r * HST;
    for (int c = q; c < De; c += 4) hr[c] = e_ln[(size_t)e * De + c];
    const float* xj = x_ln + s_src[r] * 132;
    for (int c = q; c < Dn; c += 4) hr[De + c] = xj[c];
    const float* xi = x_ln + s_dst[r] * 132;
    for (int c = q; c < Dn; c += 4) hr[De + Dn + c] = xi[c];
  }
  if (tid < 32) {
    const int Kt = De + 2 * Dn, Kp = K1steps * 4;
    for (int c = Kt; c < Kp; ++c) h[tid * HST + c] = 0.f;
  }
  __syncthreads();
  const int w = tid >> 5, lane = tid & 31;
  const int half = lane >> 4, ln = lane & 15;
  const int col = w * 16 + ln;
  // GEMM1 : h @ er1.W
  v8f c0 = {}, c1 = {};
  gemm_pair(h, HST, W1, K1steps, col, half, ln, c0, c1);
  {
    const float bb = b1[col];
#pragma unroll
    for (int r = 0; r < 8; ++r) {
      const int row = r + 8 * half;
      float v0 = c0[r] + bb; v0 = v0 >= 0.f ? v0 : 0.1f * v0;
      float v1 = c1[r] + bb; v1 = v1 >= 0.f ? v1 : 0.1f * v1;
      h2[row * H2ST + col] = v0;
      h2[(row + 16) * H2ST + col] = v1;
    }
  }
  __syncthreads();
  // GEMM2 : act @ er2.W
  v8f d0 = {}, d1 = {};
  gemm_pair(h2, H2ST, W2, 16, col, half, ln, d0, d1);
  __syncthreads();
  // gate + residual -> h2
  {
    const float bb = b2[col];
#pragma unroll
    for (int r = 0; r < 8; ++r) {
      const int row = r + 8 * half;
      const float del0 = d0[r] + bb;
      const float del1 = d1[r] + bb;
      const float g0 = 1.f / (1.f + expf(-del0));
      const float g1 = 1.f / (1.f + expf(-del1));
      float r0 = 0.f, r1 = 0.f;
      if (es_prev) {
        r0 = fmaxf(es_prev[(size_t)(e0 + row) * 64 + col], 0.f);
        r1 = fmaxf(es_prev[(size_t)(e0 + row + 16) * 64 + col], 0.f);
      }
      h2[row * H2ST + col] = r0 + g0 * del0;
      h2[(row + 16) * H2ST + col] = r1 + g1 * del1;
    }
  }
  ln_rows_lds(h2, H2ST, 64, epost_g, epost_b, tid);
  // persist e_out (pre-relu) for next layer / decoder
  for (int i = tid; i < 32 * 64; i += 128) {
    const int r = i >> 6, c = i & 63;
    es_out[(size_t)(e0 + r) * 64 + c] = h2[r * H2ST + c];
  }
  // FiLM GEMMs: S = e_out@Ws, T = e_out@Wh (shared A-frag, dual B)
  v8f s0 = {}, s1 = {}, t0 = {}, t1 = {};
  {
    const float* a0p = h2 + ln * H2ST;
    const float* a1p = h2 + (ln + 16) * H2ST;
    for (int ks = 0; ks < 16; ++ks) {
      const int k0 = ks * 4 + 2 * half;
      v2f bS; bS.x = Ws[k0 * 64 + col]; bS.y = Ws[k0 * 64 + 64 + col];
      v2f bH; bH.x = Wh[k0 * 64 + col]; bH.y = Wh[k0 * 64 + 64 + col];
      v2f a;  a.x = a0p[k0];  a.y = a0p[k0 + 1];
      s0 = wmma4(a, bS, s0); t0 = wmma4(a, bH, t0);
      v2f a1; a1.x = a1p[k0]; a1.y = a1p[k0 + 1];
      s1 = wmma4(a1, bS, s1); t1 = wmma4(a1, bH, t1);
    }
  }
  // base = x_j @ W_msg (x_j still resident in h at offset De)
  v8f p0 = {}, p1 = {};
  gemm_pair(h + De, HST, Wm, KMsteps, col, half, ln, p0, p1);
  // msg = (1 + S + bs)*base + (T + bh); scatter-add to x_aggr[dst]
  {
    const float fbs = bs[col], fbh = bh[col];
#pragma unroll
    for (int r = 0; r < 8; ++r) {
      const int row = r + 8 * half;
      const float m0 = (1.f + s0[r] + fbs) * p0[r] + (t0[r] + fbh);
      const float m1 = (1.f + s1[r] + fbs) * p1[r] + (t1[r] + fbh);
      unsafeAtomicAdd(&x_aggr[s_dst[row] * 64 + col], m0);
      unsafeAtomicAdd(&x_aggr[s_dst[row + 16] * 64 + col], m1);
    }
  }
}

// ---------------------------------------------------------------------------
// Node update: ff1 -> leaky -> ff2 (+res) -> npost LN -> relu
// ---------------------------------------------------------------------------
__global__ __launch_bounds__(128) void k_node_upd(
    const float* __restrict__ x_aggr,
    const float* __restrict__ ff1W, const float* __restrict__ ff1B,
    const float* __restrict__ ff2W, const float* __restrict__ ff2B,
    int has_res,
    const float* __restrict__ npost_g, const float* __restrict__ npost_b,
    float* x_io, float* __restrict__ xs_out) {
  __shared__ float hA[32 * H2ST];
  __shared__ float hB[32 * H2ST];
  const int tid = threadIdx.x;
  const int n0  = blockIdx.x * 32;
  for (int i = tid; i < 32 * 64; i += 128) {
    const int r = i >> 6, c = i & 63;
    hA[r * H2ST + c] = x_aggr[(n0 + r) * 64 + c];
  }
  __syncthreads();
  const int w = tid >> 5, lane = tid & 31;
  const int half = lane >> 4, ln = lane & 15;
  const int col = w * 16 + ln;
  v8f c0 = {}, c1 = {};
  gemm_pair(hA, H2ST, ff1W, 16, col, half, ln, c0, c1);
  {
    const float bb = ff1B[col];
#pragma unroll
    for (int r = 0; r < 8; ++r) {
      const int row = r + 8 * half;
      float v0 = c0[r] + bb; v0 = v0 >= 0.f ? v0 : 0.1f * v0;
      float v1 = c1[r] + bb; v1 = v1 >= 0.f ? v1 : 0.1f * v1;
      hB[row * H2ST + col] = v0;
      hB[(row + 16) * H2ST + col] = v1;
    }
  }
  __syncthreads();
  v8f d0 = {}, d1 = {};
  gemm_pair(hB, H2ST, ff2W, 16, col, half, ln, d0, d1);
  __syncthreads();
  {
    const float bb = ff2B[col];
#pragma unroll
    for (int r = 0; r < 8; ++r) {
      const int row = r + 8 * half;
      float v0 = d0[r] + bb, v1 = d1[r] + bb;
      if (has_res) {
        v0 += x_io[(n0 + row) * 132 + col];
        v1 += x_io[(n0 + row + 16) * 132 + col];
      }
      hA[row * H2ST + col] = v0;
      hA[(row + 16) * H2ST + col] = v1;
    }
  }
  ln_rows_lds(hA, H2ST, 64, npost_g, npost_b, tid);
  for (int i = tid; i < 32 * 64; i += 128) {
    const int r = i >> 6, c = i & 63;
    const float v = fmaxf(hA[r * H2ST + c], 0.f);
    x_io[(n0 + r) * 132 + c] = v;
    xs_out[(n0 + r) * 64 + c] = v;
  }
}

// ---------------------------------------------------------------------------
// JK projection: concat(xs[0..l]) @ Wjk + bjk
// ---------------------------------------------------------------------------
__global__ __launch_bounds__(128) void k_jk(const float* __restrict__ xs, int nl,
                                            const float* __restrict__ W,
                                            const float* __restrict__ bjk,
                                            float* __restrict__ x_jk) {
  __shared__ float hA[32 * JKST];
  const int tid = threadIdx.x;
  const int n0  = blockIdx.x * 32;
  const int K   = nl * 64;
  for (int i = tid; i < 32 * K; i += 128) {
    const int r = i / K, c = i - r * K;
    hA[r * JKST + c] = xs[(size_t)(c >> 6) * (NN * 64) + (n0 + r) * 64 + (c & 63)];
  }
  __syncthreads();
  const int w = tid >> 5, lane = tid & 31;
  const int half = lane >> 4, ln = lane & 15;
  const int col = w * 16 + ln;
  v8f c0 = {}, c1 = {};
  gemm_pair(hA, JKST, W, nl * 16, col, half, ln, c0, c1);
  const float bb = bjk[col];
#pragma unroll
  for (int r = 0; r < 8; ++r) {
    const int row = r + 8 * half;
    x_jk[(n0 + row) * 64 + col] = c0[r] + bb;
    x_jk[(n0 + row + 16) * 64 + col] = c1[r] + bb;
  }
}

// ---------------------------------------------------------------------------
// Decoder: LN([relu(es) | jk[src] | jk[dst]]) @ Wp + bp -> softplus -> scatter
// ---------------------------------------------------------------------------
__global__ __launch_bounds__(128) void k_dec(
    const float* __restrict__ es_l, const float* __restrict__ x_jk,
    const int* __restrict__ src, const int* __restrict__ dst,
    const float* __restrict__ dec_g, const float* __restrict__ dec_b,
    const float* __restrict__ pW, const float* __restrict__ pB,
    const int* __restrict__ winner, float* __restrict__ outP) {
  __shared__ float hA[32 * JKST];
  __shared__ int s_src[32], s_dst[32];
  const int tid = threadIdx.x;
  const int e0  = blockIdx.x * 32;
  if (tid < 32) { s_src[tid] = src[e0 + tid]; s_dst[tid] = dst[e0 + tid]; }
  __syncthreads();
  {
    const int r = tid >> 2, q = tid & 3;
    const int e = e0 + r;
    float* hr = hA + r * JKST;
    for (int c = q; c < 64; c += 4) hr[c] = fmaxf(es_l[(size_t)e * 64 + c], 0.f);
    const float* a = x_jk + s_src[r] * 64;
    for (int c = q; c < 64; c += 4) hr[64 + c] = a[c];
    const float* bptr = x_jk + s_dst[r] * 64;
    for (int c = q; c < 64; c += 4) hr[128 + c] = bptr[c];
  }
  ln_rows_lds(hA, JKST, 192, dec_g, dec_b, tid);
  const int w = tid >> 5, lane = tid & 31;
  const int half = lane >> 4, ln = lane & 15;
  const int col = w * 16 + ln;
  v8f c0 = {}, c1 = {};
  gemm_pair(hA, JKST, pW, 48, col, half, ln, c0, c1);
  const float bb = pB[col];
#pragma unroll
  for (int r = 0; r < 8; ++r) {
    const int row = r + 8 * half;
    {
      const float v  = c0[r] + bb;
      const float sp = fmaxf(v, 0.f) + log1pf(expf(-fabsf(v)));
      const int si = s_src[row], di = s_dst[row];
      if (winner[si * NN + di] == e0 + row)
        outP[((size_t)col * NN + si) * NN + di] = sp;
    }
    {
      const float v  = c1[r] + bb;
      const float sp = fmaxf(v, 0.f) + log1pf(expf(-fabsf(v)));
      const int si = s_src[row + 16], di = s_dst[row + 16];
      if (winner[si * NN + di] == e0 + row + 16)
        outP[((size_t)col * NN + si) * NN + di] = sp;
    }
  }
}

// ---------------------------------------------------------------------------
// Host orchestration
// ---------------------------------------------------------------------------
extern "C" void kernel_launch(void* const* d_in, const int* in_sizes, int n_in,
                              void* d_out, int out_size, void* d_ws,
                              size_t ws_size, hipStream_t stream) {
  (void)n_in; (void)ws_size;
  // ---- bind inputs (auto-detect pytree flattening order) ----
  const float* P0; const int* EI; const float* EA; const int *tx, *rx; int q;
  if (in_sizes[1] == 2 * EE) {              // top-level insertion order
    P0 = (const float*)d_in[0]; EI = (const int*)d_in[1];
    EA = (const float*)d_in[2];
    tx = (const int*)d_in[3];   rx = (const int*)d_in[4];
    q = 5;
  } else {                                   // top-level sorted keys
    P0 = (const float*)d_in[0]; EA = (const float*)d_in[1];
    EI = (const int*)d_in[2];
    q = 3;
    rx = (const int*)d_in[q + 73]; tx = (const int*)d_in[q + 74];
  }
  const float *nlnG[3], *nlnB[3], *elnG[3], *elnB[3], *Wmsg[3], *fsW[3], *fsB[3],
      *fhW[3], *fhB[3], *ff1W[3], *ff1B[3], *ff2W[3], *ff2B[3], *npG[3], *npB[3],
      *er1W[3], *er1B[3], *er2W[3], *er2B[3], *epG[3], *epB[3], *jkW[3], *jkB[3],
      *decG, *decB, *pW, *pB;
  auto F = [&](int i) { return (const float*)d_in[i]; };
  if (in_sizes[q] == 192) {  // params flattened with sorted dict keys
    decG = F(q); decB = F(q + 1);
    for (int l = 0; l < 3; ++l) { jkW[l] = F(q + 2 + 2 * l); jkB[l] = F(q + 3 + 2 * l); }
    const int base = q + 8;
    for (int l = 0; l < 3; ++l) {
      const int o = base + 21 * l;
      Wmsg[l] = F(o + 0);
      elnG[l] = F(o + 1);  elnB[l] = F(o + 2);
      epG[l]  = F(o + 3);  epB[l]  = F(o + 4);
      er1W[l] = F(o + 5);  er1B[l] = F(o + 6);
      er2W[l] = F(o + 7);  er2B[l] = F(o + 8);
      ff1W[l] = F(o + 9);  ff1B[l] = F(o + 10);
      ff2W[l] = F(o + 11); ff2B[l] = F(o + 12);
      fsW[l]  = F(o + 13); fsB[l]  = F(o + 14);
      fhW[l]  = F(o + 15); fhB[l]  = F(o + 16);
      nlnG[l] = F(o + 17); nlnB[l] = F(o + 18);
      npG[l]  = F(o + 19); npB[l]  = F(o + 20);
    }
    pW = F(q + 71); pB = F(q + 72);
  } else {                    // params flattened in insertion order
    for (int l = 0; l < 3; ++l) {
      const int o = q + 21 * l;
      nlnG[l] = F(o + 0);  nlnB[l] = F(o + 1);
      elnG[l] = F(o + 2);  elnB[l] = F(o + 3);
      Wmsg[l] = F(o + 4);
      fsW[l]  = F(o + 5);  fsB[l]  = F(o + 6);
      fhW[l]  = F(o + 7);  fhB[l]  = F(o + 8);
      ff1W[l] = F(o + 9);  ff1B[l] = F(o + 10);
      ff2W[l] = F(o + 11); ff2B[l] = F(o + 12);
      npG[l]  = F(o + 13); npB[l]  = F(o + 14);
      er1W[l] = F(o + 15); er1B[l] = F(o + 16);
      er2W[l] = F(o + 17); er2B[l] = F(o + 18);
      epG[l]  = F(o + 19); epB[l]  = F(o + 20);
    }
    const int jb = q + 63;
    for (int l = 0; l < 3; ++l) { jkW[l] = F(jb + 2 * l); jkB[l] = F(jb + 1 + 2 * l); }
    decG = F(q + 69); decB = F(q + 70); pW = F(q + 71); pB = F(q + 72);
  }
  // ---- workspace carve ----
  char* wp = (char*)d_ws;
  auto alloc = [&](size_t bytes) -> float* {
    float* p = (float*)wp;
    wp += (bytes + 255) & ~(size_t)255;
    return p;
  };
  float* x_cur  = alloc((size_t)NN * 132 * 4);
  float* x_ln   = alloc((size_t)NN * 132 * 4);
  float* e_ln   = alloc((size_t)EE * 128 * 4);
  float* es     = alloc((size_t)3 * EE * 64 * 4);
  float* x_aggr = alloc((size_t)NN * 64 * 4);
  float* xs     = alloc((size_t)3 * NN * 64 * 4);
  float* x_jk   = alloc((size_t)NN * 64 * 4);
  int*   winner = (int*)alloc((size_t)NN * NN * 4);
  float* w1pad  = alloc((size_t)392 * 64 * 4);
  float* wmpad  = alloc((size_t)132 * 64 * 4);

  hipMemsetAsync(d_out, 0, (size_t)out_size * 4, stream);
  hipMemsetAsync(winner, 0xFF, (size_t)NN * NN * 4, stream);
  k_winner<<<EE / 256, 256, 0, stream>>>(EI, EI + EE, winner);
  k_rowsum<<<dim3(64, NN), 256, 0, stream>>>(P0, x_cur);
  k_colsum<<<dim3(64, 3), 256, 0, stream>>>(P0, x_cur);
  k_role<<<6, 128, 0, stream>>>(tx, rx, x_cur);

  for (int l = 0; l < 3; ++l) {
    const int Dn = (l == 0) ? 131 : 64, De = (l == 0) ? 128 : 64;
    const int Kp = (l == 0) ? 392 : 192, Km = (l == 0) ? 132 : 64;
    const float *W1use, *Wmuse;
    if (l == 0) {
      k_padw<<<(392 * 64 + 255) / 256, 256, 0, stream>>>(er1W[0], 390, 392, w1pad);
      k_padw<<<(132 * 64 + 255) / 256, 256, 0, stream>>>(Wmsg[0], 131, 132, wmpad);
      W1use = w1pad; Wmuse = wmpad;
    } else {
      W1use = er1W[l]; Wmuse = Wmsg[l];
    }
    k_ln_x<<<6, 128, 0, stream>>>(x_cur, nlnG[l], nlnB[l], Dn, x_ln);
    k_ln_e<<<EE / 4, 128, 0, stream>>>(
        (l == 0) ? EA : (es + (size_t)(l - 1) * EE * 64), De, (l > 0) ? 1 : 0,
        elnG[l], elnB[l], e_ln);
    hipMemsetAsync(x_aggr, 0, (size_t)NN * 64 * 4, stream);
    k_edge<<<EE / 32, 128, 0, stream>>>(
        e_ln, De, x_ln, Dn, EI, EI + EE, W1use, Kp / 4, er1B[l], er2W[l],
        er2B[l], (l == 0) ? nullptr : (es + (size_t)(l - 1) * EE * 64), epG[l],
        epB[l], fsW[l], fsB[l], fhW[l], fhB[l], Wmuse, Km / 4,
        es + (size_t)l * EE * 64, x_aggr);
    k_node_upd<<<NN / 32, 128, 0, stream>>>(x_aggr, ff1W[l], ff1B[l], ff2W[l],
                                            ff2B[l], (l > 0) ? 1 : 0, npG[l],
                                            npB[l], x_cur,
                                            xs + (size_t)l * NN * 64);
  }
  for (int l = 0; l < 3; ++l) {
    k_jk<<<NN / 32, 128, 0, stream>>>(xs, l + 1, jkW[l], jkB[l], x_jk);
    k_dec<<<EE / 32, 128, 0, stream>>>(es + (size_t)l * EE * 64, x_jk, EI,
                                       EI + EE, decG, decB, pW, pB, winner,
                                       (float*)d_out + (size_t)l * 64 * NN * NN);
  }
}